// SpatialOutbreakSimulator_13597866459495
// MI455X (gfx1250) — compile-verified
//
#include <hip/hip_runtime.h>

typedef float v2f __attribute__((ext_vector_type(2)));
typedef float v8f __attribute__((ext_vector_type(8)));

#define ENVD 5
#define FEATD 3
#define GIN 72  /* 64 + 5 + 3 */

__device__ __forceinline__ float sigmoid_f(float x){ return 1.0f/(1.0f+expf(-x)); }
__device__ __forceinline__ float softplus_f(float x){ return x>20.0f ? x : log1pf(expf(x)); }

__device__ __forceinline__ void atomic_max_f(float* addr, float v){
  if (v >= 0.0f) atomicMax((int*)addr, __float_as_int(v));
  else           atomicMin((unsigned int*)addr, __float_as_uint(v));
}

// ------- WMMA f32 GEMM: C[M,dout] = act(A[M,K] @ W[dout,K(ldw)]^T + bias) -----------
// one wave32 per 16x(16*NT) output block; NT accumulators interleave WMMAs so the
// WMMA->WMMA accumulator RAW hazard is hidden and each A fragment feeds NT WMMAs.
template<int NT>
__global__ void gemm_wmma_f32(const float* __restrict__ A, int lda,
                              const float* __restrict__ W, int ldw,
                              const float* __restrict__ bias,
                              float* __restrict__ C, int ldc,
                              int M, int K, int dout, int relu)
{
  const int gwave = (blockIdx.x * blockDim.x + threadIdx.x) >> 5;
  const int lane  = threadIdx.x & 31;
  const int nblocks = dout / (16 * NT);
  const int mtile = gwave / nblocks;
  const int nb    = gwave - mtile * nblocks;
  const int m0 = mtile << 4;
  if (m0 >= M) return;                 // uniform per wave -> EXEC stays all-ones
  const int n0 = nb * (16 * NT);
  const int hs  = lane >> 4;           // half-wave select (K pair 0/1 vs 2/3)
  const int l15 = lane & 15;
  const float* arow = A + (size_t)(m0 + l15) * lda + (hs << 1);
  const float* wrow[NT];
  #pragma unroll
  for (int j = 0; j < NT; ++j)
    wrow[j] = W + (size_t)(n0 + 16*j + l15) * ldw + (hs << 1);
  v8f acc[NT];
  #pragma unroll
  for (int j = 0; j < NT; ++j) acc[j] = (v8f){0.f,0.f,0.f,0.f,0.f,0.f,0.f,0.f};
  for (int k = 0; k < K; k += 4) {
    v2f a; a.x = arow[k]; a.y = arow[k+1];
    #pragma unroll
    for (int j = 0; j < NT; ++j) {
      v2f b; b.x = wrow[j][k]; b.y = wrow[j][k+1];
      acc[j] = __builtin_amdgcn_wmma_f32_16x16x4_f32(false, a, false, b,
                                                     (short)0, acc[j], false, false);
    }
  }
  #pragma unroll
  for (int j = 0; j < NT; ++j) {
    const int col = n0 + 16*j + l15;
    const float bv = bias ? bias[col] : 0.0f;
    #pragma unroll
    for (int v = 0; v < 8; ++v) {
      float x = acc[j][v] + bv;
      if (relu) x = fmaxf(x, 0.0f);
      C[(size_t)(m0 + (hs<<3) + v) * ldc + col] = x;
    }
  }
}

// --------- one-time: pre_e[e] = b1a + attnW1[:, :64] @ (edge_enc MLP(edge_attr)) ------
__global__ void edge_precompute(const float* __restrict__ eattr,
                                const float* __restrict__ eW1, const float* __restrict__ eb1,
                                const float* __restrict__ eW2, const float* __restrict__ eb2,
                                const float* __restrict__ aW1, const float* __restrict__ ab1,
                                float* __restrict__ pre_e, int E)
{
  const int e    = (blockIdx.x * blockDim.x + threadIdx.x) >> 5;
  const int lane = threadIdx.x & 31;
  if (e >= E) return;
  const float a0 = eattr[(size_t)e*4+0], a1 = eattr[(size_t)e*4+1];
  const float a2 = eattr[(size_t)e*4+2], a3 = eattr[(size_t)e*4+3];
  const int j0 = lane, j1 = lane + 32;
  float h1a = eb1[j0] + a0*eW1[j0*4+0]+a1*eW1[j0*4+1]+a2*eW1[j0*4+2]+a3*eW1[j0*4+3];
  float h1b = eb1[j1] + a0*eW1[j1*4+0]+a1*eW1[j1*4+1]+a2*eW1[j1*4+2]+a3*eW1[j1*4+3];
  h1a = fmaxf(h1a, 0.f); h1b = fmaxf(h1b, 0.f);
  float h2a = eb2[j0], h2b = eb2[j1];
  for (int k = 0; k < 64; ++k) {
    float hk = __shfl(k < 32 ? h1a : h1b, k & 31, 32);
    h2a += hk * eW2[j0*64+k];
    h2b += hk * eW2[j1*64+k];
  }
  float pa = ab1[j0], pb = ab1[j1];
  for (int k = 0; k < 64; ++k) {
    float hk = __shfl(k < 32 ? h2a : h2b, k & 31, 32);
    pa += hk * aW1[j0*128+k];     // attn W1 first-64 columns (edge part), ldw=128
    pb += hk * aW1[j1*128+k];
  }
  pre_e[(size_t)e*64 + j0] = pa;
  pre_e[(size_t)e*64 + j1] = pb;
}

__global__ void init_state(float* __restrict__ h, float* __restrict__ lice,
                           const float* __restrict__ lice0, int N)
{
  int i = blockIdx.x*blockDim.x + threadIdx.x;
  if (i < N*64) h[i] = 0.f;
  if (i < N*3)  lice[i] = lice0[i];
}

__global__ void clear_step(float* __restrict__ pressure, float* __restrict__ segmax,
                           float* __restrict__ segsum, int N)
{
  int i = blockIdx.x*blockDim.x + threadIdx.x;
  if (i < N*64) pressure[i] = 0.f;
  if (i < N) { segmax[i] = -__builtin_inff(); segsum[i] = 0.f; }
}

// fecundity MLP (1->32->32->1 softplus) + env_enc layer1 (5->64 relu)
__global__ void node_fec_env1(const float* __restrict__ env_t,
                              const float* fW1, const float* fb1,
                              const float* fW2, const float* fb2,
                              const float* fW3, const float* fb3,
                              const float* eW1, const float* eb1,
                              float* __restrict__ fec, float* __restrict__ env_h1, int N)
{
  int i = blockIdx.x*blockDim.x + threadIdx.x;
  if (i >= N) return;
  const float* ev = env_t + (size_t)i*ENVD;
  float tn = (ev[0] - 10.0f) * 0.2f;
  float x1[32];
  #pragma unroll
  for (int j=0;j<32;++j) x1[j] = fmaxf(fb1[j] + fW1[j]*tn, 0.f);
  float x2[32];
  #pragma unroll
  for (int j=0;j<32;++j){
    float s = fb2[j];
    #pragma unroll
    for (int k=0;k<32;++k) s += x1[k]*fW2[j*32+k];
    x2[j] = fmaxf(s, 0.f);
  }
  float y = fb3[0];
  #pragma unroll
  for (int k=0;k<32;++k) y += x2[k]*fW3[k];
  fec[i] = softplus_f(y);
  float e0=ev[0],e1=ev[1],e2=ev[2],e3=ev[3],e4=ev[4];
  for (int j=0;j<64;++j){
    float s = eb1[j] + e0*eW1[j*5+0]+e1*eW1[j*5+1]+e2*eW1[j*5+2]+e3*eW1[j*5+3]+e4*eW1[j*5+4];
    env_h1[(size_t)i*64+j] = fmaxf(s, 0.f);
  }
}

// pass 1: logit[e] = w2 . relu(pre_e[e] + Hw[src]) + b2 ; segment max over dst
__global__ void edge_logits(const float* __restrict__ pre_e, const float* __restrict__ Hw,
                            const int* __restrict__ src, const int* __restrict__ dst,
                            const float* __restrict__ aW2, const float* __restrict__ ab2,
                            float* __restrict__ logits, float* __restrict__ segmax, int E)
{
  int e    = (blockIdx.x*blockDim.x + threadIdx.x) >> 5;
  int lane = threadIdx.x & 31;
  if (e >= E) return;
  int s = src[e];
  const float* pe = pre_e + (size_t)e*64;
  const float* hw = Hw    + (size_t)s*64;
  float p = fmaxf(pe[lane]      + hw[lane],      0.f) * aW2[lane]
          + fmaxf(pe[lane + 32] + hw[lane + 32], 0.f) * aW2[lane + 32];
  #pragma unroll
  for (int off = 16; off > 0; off >>= 1) p += __shfl_xor(p, off, 32);
  if (lane == 0) {
    float lg = p + ab2[0];
    logits[e] = lg;
    atomic_max_f(&segmax[dst[e]], lg);
  }
}

// pass 2 (fused): ex = exp(logit - fixed_max); segsum += ex;
// pressure_un[dst] += ex * lice[src,0] * beta * h[src]   (normalized later per node)
__global__ void edge_accum(const int* __restrict__ src, const int* __restrict__ dst,
                           const float* __restrict__ logits, const float* __restrict__ segmax,
                           const float* __restrict__ lice, const float* __restrict__ h,
                           const float* __restrict__ log_beta,
                           float* __restrict__ segsum, float* __restrict__ pressure, int E)
{
  int e    = (blockIdx.x*blockDim.x + threadIdx.x) >> 5;
  int lane = threadIdx.x & 31;
  if (e >= E) return;
  int s = src[e], d = dst[e];
  float m = segmax[d];
  if (!(m >= -3.0e38f)) m = 0.f;              // -inf (no in-edges) / NaN -> 0
  float ex = expf(logits[e] - m);
  float c  = ex * lice[(size_t)s*3] * expf(log_beta[0]);
  const float* hs = h + (size_t)s*64;
  if (lane == 0) atomicAdd(&segsum[d], ex);
  atomicAdd(&pressure[(size_t)d*64 + lane],      c * hs[lane]);
  atomicAdd(&pressure[(size_t)d*64 + lane + 32], c * hs[lane + 32]);
}

// dynin = [ enc + pressure_un/(segsum+1e-8) | env_t | lice ]
__global__ void assemble_dynin(const float* __restrict__ enc, const float* __restrict__ pressure,
                               const float* __restrict__ segsum,
                               const float* __restrict__ env_t, const float* __restrict__ lice,
                               float* __restrict__ dynin, int N)
{
  int idx = blockIdx.x*blockDim.x + threadIdx.x;
  if (idx >= N*GIN) return;
  int i = idx / GIN, j = idx - i*GIN;
  float v;
  if (j < 64) {
    float S = segsum[i] + 1e-8f;
    v = enc[(size_t)i*64 + j] + pressure[(size_t)i*64 + j] / S;
  }
  else if (j < 69) v = env_t[(size_t)i*ENVD + (j - 64)];
  else             v = lice[(size_t)i*FEATD + (j - 69)];
  dynin[idx] = v;
}

__global__ void gru_gate(const float* __restrict__ gi, const float* __restrict__ gh,
                         const float* __restrict__ h, float* __restrict__ hnew, int N)
{
  int idx = blockIdx.x*blockDim.x + threadIdx.x;
  if (idx >= N*64) return;
  int i = idx >> 6, j = idx & 63;
  const float* gii = gi + (size_t)i*192;
  const float* ghh = gh + (size_t)i*192;
  float r = sigmoid_f(gii[j]       + ghh[j]);
  float z = sigmoid_f(gii[64 + j]  + ghh[64 + j]);
  float n = tanhf(gii[128 + j] + r * ghh[128 + j]);
  hnew[idx] = (1.f - z) * n + z * h[idx];
}

// dec layer2 (64->3) + softplus + temp scaling; writes lice_new and traj[t]
__global__ void node_out(const float* __restrict__ d1,
                         const float* dW2, const float* db2,
                         const float* __restrict__ fec, const float* temp_sens,
                         float* __restrict__ lice_new, float* __restrict__ traj_t, int N)
{
  int i = blockIdx.x*blockDim.x + threadIdx.x;
  if (i >= N) return;
  const float* x = d1 + (size_t)i*64;
  float fac = 1.0f + temp_sens[0] * (fec[i] - 1.0f);
  #pragma unroll
  for (int c = 0; c < 3; ++c){
    float s = db2[c];
    for (int k = 0; k < 64; ++k) s += x[k] * dW2[c*64 + k];
    float v = softplus_f(s) * fac;
    lice_new[(size_t)i*3 + c] = v;
    traj_t[(size_t)i*3 + c]   = v;
  }
}

extern "C" void kernel_launch(void* const* d_in, const int* in_sizes, int n_in,
                              void* d_out, int out_size, void* d_ws, size_t ws_size,
                              hipStream_t stream)
{
  // setup_inputs order, params flattened in pytree (sorted-key) order
  const float* env_seq   = (const float*)d_in[0];
  const int*   edge_idx  = (const int*)  d_in[1];
  const float* edge_attr = (const float*)d_in[2];
  const float* lice0     = (const float*)d_in[3];
  const float* aW1 = (const float*)d_in[4];   const float* ab1 = (const float*)d_in[5];
  const float* aW2 = (const float*)d_in[6];   const float* ab2 = (const float*)d_in[7];
  const float* dW1 = (const float*)d_in[8];   const float* db1 = (const float*)d_in[9];
  const float* dW2 = (const float*)d_in[10];  const float* db2 = (const float*)d_in[11];
  const float* eW1 = (const float*)d_in[12];  const float* eb1 = (const float*)d_in[13];
  const float* eW2 = (const float*)d_in[14];  const float* eb2 = (const float*)d_in[15];
  const float* nW1 = (const float*)d_in[16];  const float* nb1 = (const float*)d_in[17];
  const float* nW2 = (const float*)d_in[18];  const float* nb2 = (const float*)d_in[19];
  const float* nW3 = (const float*)d_in[20];  const float* nb3 = (const float*)d_in[21];
  const float* fW1 = (const float*)d_in[22];  const float* fb1 = (const float*)d_in[23];
  const float* fW2 = (const float*)d_in[24];  const float* fb2 = (const float*)d_in[25];
  const float* fW3 = (const float*)d_in[26];  const float* fb3 = (const float*)d_in[27];
  const float* gbhh = (const float*)d_in[28];
  const float* gbih = (const float*)d_in[29];
  const float* gWhh = (const float*)d_in[30];
  const float* gWih = (const float*)d_in[31];
  const float* log_beta  = (const float*)d_in[32];
  const float* temp_sens = (const float*)d_in[33];

  const int E = in_sizes[2] / 4;
  const int N = in_sizes[3] / 3;
  const int T = in_sizes[0] / (N * ENVD);

  const int* src = edge_idx;
  const int* dst = edge_idx + E;

  float* w = (float*)d_ws;
  size_t off = 0;
  auto alloc = [&](size_t n){ float* p = w + off; off += (n + 63) & ~(size_t)63; return p; };
  float* pre_e    = alloc((size_t)E * 64);
  float* hbuf0    = alloc((size_t)N * 64);
  float* hbuf1    = alloc((size_t)N * 64);
  float* lbuf0    = alloc((size_t)N * 3);
  float* lbuf1    = alloc((size_t)N * 3);
  float* tmpA     = alloc((size_t)N * 64);
  float* tmpB     = alloc((size_t)N * 64);
  float* enc      = alloc((size_t)N * 64);
  float* Hw       = alloc((size_t)N * 64);
  float* pressure = alloc((size_t)N * 64);
  float* dynin    = alloc((size_t)N * GIN);
  float* gi       = alloc((size_t)N * 192);
  float* gh       = alloc((size_t)N * 192);
  float* fec      = alloc((size_t)N);
  float* segmax   = alloc((size_t)N);
  float* segsum   = alloc((size_t)N);
  float* logits   = alloc((size_t)E);
  (void)ws_size; (void)n_in; (void)out_size;

  const int B = 256;
  auto nblk = [&](long long t){ return (unsigned)((t + B - 1) / B); };

  // time-invariant edge path, hoisted out of the scan
  edge_precompute<<<nblk((long long)E * 32), B, 0, stream>>>(edge_attr, eW1, eb1, eW2, eb2,
                                                             aW1, ab1, pre_e, E);
  init_state<<<nblk((long long)N * 64), B, 0, stream>>>(hbuf0, lbuf0, lice0, N);

  auto gemm = [&](const float* A, int lda, const float* W, int ldw, const float* bias,
                  float* C, int ldc, int M, int K, int dout, int relu){
    long long waves = (long long)((M + 15) / 16) * (dout / 64);   // NT=4 -> 64 cols/wave
    gemm_wmma_f32<4><<<nblk(waves * 32), B, 0, stream>>>(A, lda, W, ldw, bias,
                                                         C, ldc, M, K, dout, relu);
  };

  float* hcur = hbuf0; float* hnext = hbuf1;
  float* lcur = lbuf0; float* lnext = lbuf1;
  float* traj = (float*)d_out;

  for (int t = 0; t < T; ++t) {
    const float* env_t = env_seq + (size_t)t * N * ENVD;
    clear_step<<<nblk((long long)N * 64), B, 0, stream>>>(pressure, segmax, segsum, N);
    node_fec_env1<<<nblk(N), B, 0, stream>>>(env_t, fW1, fb1, fW2, fb2, fW3, fb3,
                                             nW1, nb1, fec, tmpA, N);
    gemm(tmpA, 64, nW2, 64, nb2, tmpB, 64, N, 64, 64, 1);          // env_enc L2 (relu)
    gemm(tmpB, 64, nW3, 64, nb3, enc,  64, N, 64, 64, 0);          // env_enc L3
    gemm(hcur, 64, aW1 + 64, 128, nullptr, Hw, 64, N, 64, 64, 0);  // h @ W1h^T (attn split)
    edge_logits<<<nblk((long long)E * 32), B, 0, stream>>>(pre_e, Hw, src, dst, aW2, ab2,
                                                           logits, segmax, E);
    edge_accum<<<nblk((long long)E * 32), B, 0, stream>>>(src, dst, logits, segmax,
                                                          lcur, hcur, log_beta,
                                                          segsum, pressure, E);
    assemble_dynin<<<nblk((long long)N * GIN), B, 0, stream>>>(enc, pressure, segsum,
                                                               env_t, lcur, dynin, N);
    gemm(dynin, GIN, gWih, GIN, gbih, gi, 192, N, GIN, 192, 0);    // GRU input gates
    gemm(hcur, 64, gWhh, 64, gbhh, gh, 192, N, 64, 192, 0);        // GRU hidden gates
    gru_gate<<<nblk((long long)N * 64), B, 0, stream>>>(gi, gh, hcur, hnext, N);
    gemm(hnext, 64, dW1, 64, db1, tmpA, 64, N, 64, 64, 1);         // dec L1 (relu)
    node_out<<<nblk(N), B, 0, stream>>>(tmpA, dW2, db2, fec, temp_sens,
                                        lnext, traj + (size_t)t * N * 3, N);
    float* th = hcur; hcur = hnext; hnext = th;
    float* tl = lcur; lcur = lnext; lnext = tl;
  }
}